// CESAR_24309514895978
// MI455X (gfx1250) — compile-verified
//
#include <hip/hip_runtime.h>
#include <hip/hip_bf16.h>
#include <math.h>

// ---------------------------------------------------------------------------
// B=16, S=512, D=1024.
//   cs[b] = (sum_{valid} exp(l-M) * |cos|) / (sum_{valid} exp(l-M))
// All GEMM-like ops: V_WMMA_F32_16X16X4_F32 (f32 precision is free: problem
// fits in the 192MB L2, so it's bandwidth-bound, not WMMA-rate bound).
// CDNA5 data movement: B-tiles staged to LDS with GLOBAL_LOAD_ASYNC_TO_LDS_B128
// (ASYNCcnt + s_wait_asynccnt), double-buffered so the async DMA of chunk c+1
// overlaps the WMMA loop on chunk c; fragments then come from LDS (ds_load_b64).
// ---------------------------------------------------------------------------

#define B_   16
#define S_   512
#define D_   1024
#define BS_  (B_ * S_)
#define QKLD 2048
#define NEGV (-1e30f)
#define EPSN (1e-12f)

#define KC   32                 // k-chunk staged per step
#define NC   (D_ / KC)          // 32 chunks
#define LSTR 36                 // LDS row stride (floats): 144B rows, 16B aligned,
                                // stride-4 bank pattern -> conflict-free b64 reads

typedef __attribute__((ext_vector_type(2))) float v2f;
typedef __attribute__((ext_vector_type(8))) float v8f;

__device__ __forceinline__ v2f ld2(const float* p) { return *(const v2f*)p; }

__device__ __forceinline__ v8f wmma_f32(v2f a, v2f b, v8f c) {
  // (neg_a, A, neg_b, B, c_mod, C, reuse_a, reuse_b)
  return __builtin_amdgcn_wmma_f32_16x16x4_f32(false, a, false, b, (short)0, c,
                                               false, false);
}

__device__ __forceinline__ float wave_max(float v) {
  #pragma unroll
  for (int m = 16; m; m >>= 1) v = fmaxf(v, __shfl_xor(v, m, 32));
  return v;
}
__device__ __forceinline__ float wave_sum(float v) {
  #pragma unroll
  for (int m = 16; m; m >>= 1) v += __shfl_xor(v, m, 32);
  return v;
}

// Wait for this wave's outstanding async (LDS-DMA) loads.
__device__ __forceinline__ void wait_async0() {
#if __has_builtin(__builtin_amdgcn_s_wait_asynccnt)
  __builtin_amdgcn_s_wait_asynccnt(0);
#else
  asm volatile("s_wait_asynccnt 0x0" ::: "memory");
#endif
}

// Stage a 64-row x KC-float tile (row-major src, leading dim `ld`) into LDS
// at `lbase` (row stride LSTR floats) using async b128 global->LDS DMA.
// 128 threads: wave w copies rows w*16..w*16+15; each lane moves 16B.
__device__ __forceinline__ void async_stage(float* lbase,
                                            const float* src, int ld, int tid) {
  const int wave = tid >> 5, lane = tid & 31;
  const int seg  = lane & 7;     // 16B segment within a 128B row
  const int rsub = lane >> 3;    // 0..3
  const uint32_t lds0 = (uint32_t)(uintptr_t)lbase;  // LDS byte offset (low 32b)
  #pragma unroll
  for (int i = 0; i < 4; ++i) {
    const int row = wave * 16 + i * 4 + rsub;
    const uint32_t laddr = lds0 + (uint32_t)(row * LSTR + seg * 4) * 4u;
    const float* g = src + (size_t)row * ld + seg * 4;
    asm volatile("global_load_async_to_lds_b128 %0, %1, off"
                 :: "v"(laddr), "v"(g) : "memory");
  }
}

__device__ __forceinline__ v2f lds_frag(const float* buf, int n, int koff) {
  return *(const v2f*)(buf + n * LSTR + koff);
}

// ---------------------------------------------------------------------------
// K1: QK[r,0:1024] = E @ Wq^T + bq ; QK[r,1024:2048] = E @ Wk^T + bk.
// Block = 64 rows x 64 cols; B-tile (weights) staged in LDS, shared by 4 waves.
// ---------------------------------------------------------------------------
__global__ void __launch_bounds__(128)
cesar_qk_gemm(const float* __restrict__ E,  const float* __restrict__ Wq,
              const float* __restrict__ bq, const float* __restrict__ Wk,
              const float* __restrict__ bk, float* __restrict__ QK) {
  __shared__ __attribute__((aligned(16))) float sB[2][64][LSTR];
  const int tid = threadIdx.x, lane = tid & 31, wave = tid >> 5;
  const int mgrp = blockIdx.x >> 5;            // 0..127
  const int ngrp = blockIdx.x & 31;            // 0..31
  const int col_base = ngrp * 64;
  const bool isQ = (col_base < D_);
  const float* W    = isQ ? Wq : Wk;
  const float* bias = isQ ? bq : bk;
  const int ncol0 = isQ ? col_base : col_base - D_;
  const int row_base = mgrp * 64 + wave * 16;
  const int lh = lane >> 4, ll = lane & 15;

  const float* arow = E + (size_t)(row_base + ll) * D_ + lh * 2;
  const float* bsrc = W + (size_t)ncol0 * D_;   // 64 weight rows, ld = D_

  v8f c0 = {}, c1 = {}, c2 = {}, c3 = {};
  async_stage(&sB[0][0][0], bsrc, D_, tid);
  for (int c = 0; c < NC; ++c) {
    wait_async0();
    __syncthreads();
    if (c + 1 < NC) async_stage(&sB[(c + 1) & 1][0][0], bsrc + (c + 1) * KC, D_, tid);
    const float* bb = &sB[c & 1][0][0];
    const int kc = c * KC;
    #pragma unroll
    for (int kk = 0; kk < KC; kk += 4) {
      v2f a  = ld2(arow + kc + kk);
      v2f b0 = lds_frag(bb, ll,      kk + lh * 2);
      v2f b1 = lds_frag(bb, 16 + ll, kk + lh * 2);
      v2f b2 = lds_frag(bb, 32 + ll, kk + lh * 2);
      v2f b3 = lds_frag(bb, 48 + ll, kk + lh * 2);
      c0 = wmma_f32(a, b0, c0);
      c1 = wmma_f32(a, b1, c1);
      c2 = wmma_f32(a, b2, c2);
      c3 = wmma_f32(a, b3, c3);
    }
  }

  const float bv0 = bias[ncol0 + ll];
  const float bv1 = bias[ncol0 + 16 + ll];
  const float bv2 = bias[ncol0 + 32 + ll];
  const float bv3 = bias[ncol0 + 48 + ll];
  #pragma unroll
  for (int r = 0; r < 8; ++r) {
    const int row = row_base + r + lh * 8;     // C/D layout: M = r (+8 hi half)
    float* o = QK + (size_t)row * QKLD + col_base + ll;
    o[0]  = c0[r] + bv0;
    o[16] = c1[r] + bv1;
    o[32] = c2[r] + bv2;
    o[48] = c3[r] + bv3;
  }
}

// ---------------------------------------------------------------------------
// K2: nrm[r] = max(||E[r,:]||, 1e-12).  One wave per row.
// ---------------------------------------------------------------------------
__global__ void cesar_row_norms(const float* __restrict__ E,
                                float* __restrict__ nrm) {
  const int row  = blockIdx.x * 8 + (threadIdx.x >> 5);
  const int lane = threadIdx.x & 31;
  const float* p = E + (size_t)row * D_;
  float s = 0.f;
  #pragma unroll 4
  for (int d = lane; d < D_; d += 32) { const float v = p[d]; s += v * v; }
  s = wave_sum(s);
  if (lane == 0) nrm[row] = fmaxf(sqrtf(s), EPSN);
}

// ---------------------------------------------------------------------------
// K3: pass 1 — block = batch b, 64 i-rows x 64 j-cols; K-rows staged in LDS.
// Per-wave-tile max of masked logits -> maxpart[wt].
// ---------------------------------------------------------------------------
__global__ void __launch_bounds__(128)
cesar_attn_max(const float* __restrict__ QK, const int* __restrict__ am,
               const int* __restrict__ tt, float* __restrict__ maxpart) {
  __shared__ __attribute__((aligned(16))) float sB[2][64][LSTR];
  const int tid = threadIdx.x, lane = tid & 31, wave = tid >> 5;
  const int b    = blockIdx.x >> 6;            // 16 batches
  const int rem  = blockIdx.x & 63;
  const int ibase = (rem >> 3) * 64 + wave * 16;
  const int jbase = (rem & 7) * 64;
  const int lh = lane >> 4, ll = lane & 15;

  const float* qrow = QK + (size_t)(b * S_ + ibase + ll) * QKLD + lh * 2;
  const float* ksrc = QK + (size_t)(b * S_ + jbase) * QKLD + D_;  // 64 K rows

  v8f c0 = {}, c1 = {}, c2 = {}, c3 = {};
  async_stage(&sB[0][0][0], ksrc, QKLD, tid);
  for (int c = 0; c < NC; ++c) {
    wait_async0();
    __syncthreads();
    if (c + 1 < NC) async_stage(&sB[(c + 1) & 1][0][0], ksrc + (c + 1) * KC, QKLD, tid);
    const float* bb = &sB[c & 1][0][0];
    const int kc = c * KC;
    #pragma unroll
    for (int kk = 0; kk < KC; kk += 4) {
      v2f a  = ld2(qrow + kc + kk);
      v2f b0 = lds_frag(bb, ll,      kk + lh * 2);
      v2f b1 = lds_frag(bb, 16 + ll, kk + lh * 2);
      v2f b2 = lds_frag(bb, 32 + ll, kk + lh * 2);
      v2f b3 = lds_frag(bb, 48 + ll, kk + lh * 2);
      c0 = wmma_f32(a, b0, c0);
      c1 = wmma_f32(a, b1, c1);
      c2 = wmma_f32(a, b2, c2);
      c3 = wmma_f32(a, b3, c3);
    }
  }

  const int* amb = am + b * S_;
  const int* ttb = tt + b * S_;
  const bool v1_0 = (amb[jbase + ll]      == 1) && (ttb[jbase + ll]      == 1);
  const bool v1_1 = (amb[jbase + 16 + ll] == 1) && (ttb[jbase + 16 + ll] == 1);
  const bool v1_2 = (amb[jbase + 32 + ll] == 1) && (ttb[jbase + 32 + ll] == 1);
  const bool v1_3 = (amb[jbase + 48 + ll] == 1) && (ttb[jbase + 48 + ll] == 1);

  float mx = NEGV;
  #pragma unroll
  for (int r = 0; r < 8; ++r) {
    const int i = ibase + r + lh * 8;
    const bool v0 = (amb[i] == 1) && (ttb[i] == 0);
    mx = fmaxf(mx, (v0 && v1_0) ? c0[r] : NEGV);
    mx = fmaxf(mx, (v0 && v1_1) ? c1[r] : NEGV);
    mx = fmaxf(mx, (v0 && v1_2) ? c2[r] : NEGV);
    mx = fmaxf(mx, (v0 && v1_3) ? c3[r] : NEGV);
  }
  mx = wave_max(mx);
  if (lane == 0) maxpart[blockIdx.x * 4 + wave] = mx;
}

// K4: reduce 256 tile maxima per batch -> Mb[b].  Fixed order, deterministic.
__global__ void cesar_reduce_max(const float* __restrict__ maxpart,
                                 float* __restrict__ Mb) {
  __shared__ float sm[8];
  const int b = blockIdx.x, t = threadIdx.x;
  float v = maxpart[b * 256 + t];
  v = wave_max(v);
  if ((t & 31) == 0) sm[t >> 5] = v;
  __syncthreads();
  if (t == 0) {
    float r = sm[0];
    #pragma unroll
    for (int i = 1; i < 8; ++i) r = fmaxf(r, sm[i]);
    Mb[b] = r;
  }
}

// ---------------------------------------------------------------------------
// K5: pass 2 — logits + Gram tiles (8 WMMAs / k-step); both B-tiles (K-rows
// and E-rows) staged via async LDS DMA, double-buffered.
// ---------------------------------------------------------------------------
__global__ void __launch_bounds__(128)
cesar_attn_sum(const float* __restrict__ QK, const float* __restrict__ E,
               const int* __restrict__ am, const int* __restrict__ tt,
               const float* __restrict__ nrm, const float* __restrict__ Mb,
               float* __restrict__ swpart, float* __restrict__ swspart) {
  __shared__ __attribute__((aligned(16))) float sK[2][64][LSTR];
  __shared__ __attribute__((aligned(16))) float sE[2][64][LSTR];
  const int tid = threadIdx.x, lane = tid & 31, wave = tid >> 5;
  const int b    = blockIdx.x >> 6;
  const int rem  = blockIdx.x & 63;
  const int ibase = (rem >> 3) * 64 + wave * 16;
  const int jbase = (rem & 7) * 64;
  const int lh = lane >> 4, ll = lane & 15;

  const float* qrow = QK + (size_t)(b * S_ + ibase + ll) * QKLD + lh * 2;
  const float* eirw = E  + (size_t)(b * S_ + ibase + ll) * D_   + lh * 2;
  const float* ksrc = QK + (size_t)(b * S_ + jbase) * QKLD + D_;
  const float* esrc = E  + (size_t)(b * S_ + jbase) * D_;

  v8f c0 = {}, c1 = {}, c2 = {}, c3 = {};   // logits
  v8f g0 = {}, g1 = {}, g2 = {}, g3 = {};   // gram
  async_stage(&sK[0][0][0], ksrc, QKLD, tid);
  async_stage(&sE[0][0][0], esrc, D_,   tid);
  for (int c = 0; c < NC; ++c) {
    wait_async0();
    __syncthreads();
    if (c + 1 < NC) {
      async_stage(&sK[(c + 1) & 1][0][0], ksrc + (c + 1) * KC, QKLD, tid);
      async_stage(&sE[(c + 1) & 1][0][0], esrc + (c + 1) * KC, D_,   tid);
    }
    const float* bk_ = &sK[c & 1][0][0];
    const float* be_ = &sE[c & 1][0][0];
    const int kc = c * KC;
    #pragma unroll
    for (int kk = 0; kk < KC; kk += 4) {
      v2f aq = ld2(qrow + kc + kk);
      v2f ae = ld2(eirw + kc + kk);
      v2f b0 = lds_frag(bk_, ll,      kk + lh * 2);
      v2f b1 = lds_frag(bk_, 16 + ll, kk + lh * 2);
      v2f b2 = lds_frag(bk_, 32 + ll, kk + lh * 2);
      v2f b3 = lds_frag(bk_, 48 + ll, kk + lh * 2);
      v2f e0 = lds_frag(be_, ll,      kk + lh * 2);
      v2f e1 = lds_frag(be_, 16 + ll, kk + lh * 2);
      v2f e2 = lds_frag(be_, 32 + ll, kk + lh * 2);
      v2f e3 = lds_frag(be_, 48 + ll, kk + lh * 2);
      c0 = wmma_f32(aq, b0, c0);
      c1 = wmma_f32(aq, b1, c1);
      c2 = wmma_f32(aq, b2, c2);
      c3 = wmma_f32(aq, b3, c3);
      g0 = wmma_f32(ae, e0, g0);
      g1 = wmma_f32(ae, e1, g1);
      g2 = wmma_f32(ae, e2, g2);
      g3 = wmma_f32(ae, e3, g3);
    }
  }

  const int* amb = am + b * S_;
  const int* ttb = tt + b * S_;
  const bool v1_0 = (amb[jbase + ll]      == 1) && (ttb[jbase + ll]      == 1);
  const bool v1_1 = (amb[jbase + 16 + ll] == 1) && (ttb[jbase + 16 + ll] == 1);
  const bool v1_2 = (amb[jbase + 32 + ll] == 1) && (ttb[jbase + 32 + ll] == 1);
  const bool v1_3 = (amb[jbase + 48 + ll] == 1) && (ttb[jbase + 48 + ll] == 1);
  const float njr0 = 1.f / nrm[b * S_ + jbase + ll];
  const float njr1 = 1.f / nrm[b * S_ + jbase + 16 + ll];
  const float njr2 = 1.f / nrm[b * S_ + jbase + 32 + ll];
  const float njr3 = 1.f / nrm[b * S_ + jbase + 48 + ll];
  const float M = Mb[b];

  float sw = 0.f, sws = 0.f;
  #pragma unroll
  for (int r = 0; r < 8; ++r) {
    const int i = ibase + r + lh * 8;
    const bool v0 = (amb[i] == 1) && (ttb[i] == 0);
    const float nir = 1.f / nrm[b * S_ + i];
    if (v0 && v1_0) { const float w = __expf(c0[r] - M); sw += w; sws += w * fabsf(g0[r]) * nir * njr0; }
    if (v0 && v1_1) { const float w = __expf(c1[r] - M); sw += w; sws += w * fabsf(g1[r]) * nir * njr1; }
    if (v0 && v1_2) { const float w = __expf(c2[r] - M); sw += w; sws += w * fabsf(g2[r]) * nir * njr2; }
    if (v0 && v1_3) { const float w = __expf(c3[r] - M); sw += w; sws += w * fabsf(g3[r]) * nir * njr3; }
  }
  sw  = wave_sum(sw);
  sws = wave_sum(sws);
  if (lane == 0) {
    swpart[blockIdx.x * 4 + wave]  = sw;
    swspart[blockIdx.x * 4 + wave] = sws;
  }
}

// K6: cs[b] = sws / sw (0 if no valid pairs).  Fixed-order sums.
__global__ void cesar_finalize(const float* __restrict__ swpart,
                               const float* __restrict__ swspart,
                               float* __restrict__ out) {
  __shared__ float s1[8], s2[8];
  const int b = blockIdx.x, t = threadIdx.x;
  float a = swpart[b * 256 + t];
  float c = swspart[b * 256 + t];
  a = wave_sum(a);
  c = wave_sum(c);
  if ((t & 31) == 0) { s1[t >> 5] = a; s2[t >> 5] = c; }
  __syncthreads();
  if (t == 0) {
    float sw = 0.f, sws = 0.f;
    #pragma unroll
    for (int i = 0; i < 8; ++i) { sw += s1[i]; sws += s2[i]; }
    out[b] = (sw > 0.f) ? (sws / sw) : 0.f;
  }
}

// ---------------------------------------------------------------------------
extern "C" void kernel_launch(void* const* d_in, const int* in_sizes, int n_in,
                              void* d_out, int out_size, void* d_ws,
                              size_t ws_size, hipStream_t stream) {
  (void)in_sizes; (void)n_in; (void)out_size; (void)ws_size;
  const float* E  = (const float*)d_in[0];
  const float* Wq = (const float*)d_in[1];
  const float* bq = (const float*)d_in[2];
  const float* Wk = (const float*)d_in[3];
  const float* bk = (const float*)d_in[4];
  const int*   am = (const int*)d_in[5];
  const int*   tt = (const int*)d_in[6];
  float* out = (float*)d_out;

  // Workspace layout (floats): ~67.2 MB.
  float* ws      = (float*)d_ws;
  float* QK      = ws;                              // 8192*2048
  float* nrm     = QK + (size_t)BS_ * QKLD;         // 8192
  float* maxpart = nrm + BS_;                       // 4096
  float* Mb      = maxpart + 4096;                  // 16
  float* swp     = Mb + 16;                         // 4096
  float* swsp    = swp + 4096;                      // 4096

  cesar_qk_gemm   <<<4096, 128, 0, stream>>>(E, Wq, bq, Wk, bk, QK);
  cesar_row_norms <<<1024, 256, 0, stream>>>(E, nrm);
  cesar_attn_max  <<<1024, 128, 0, stream>>>(QK, am, tt, maxpart);
  cesar_reduce_max<<<16, 256, 0, stream>>>(maxpart, Mb);
  cesar_attn_sum  <<<1024, 128, 0, stream>>>(QK, E, am, tt, nrm, Mb, swp, swsp);
  cesar_finalize  <<<16, 256, 0, stream>>>(swp, swsp, out);
}